// InductionHeadAttentionSmaller_1778116461294
// MI455X (gfx1250) — compile-verified
//
#include <hip/hip_runtime.h>
#include <cmath>

#define SD 2048   // S == V == 2048
#define BB 16

typedef __attribute__((ext_vector_type(2))) float v2f;
typedef __attribute__((ext_vector_type(8))) float v8f;

// ---------------------------------------------------------------------------
// C[2048x2048] = A @ B^T  (both row-major 2048x2048), f32 WMMA 16x16x4.
// 64x64 macro-tile per 256-thread block (8 wave32 waves); each wave computes a
// 16x32 strip (two 16x16 C tiles sharing one A fragment).
// ---------------------------------------------------------------------------
__global__ __launch_bounds__(256) void gemm_nt_wmma_f32(
    const float* __restrict__ A, const float* __restrict__ B,
    float* __restrict__ C) {
  constexpr int LP = 36;                       // LDS row pitch (16B-aligned, odd/4 -> conflict-free)
  __shared__ float As[64 * LP];
  __shared__ float Bs[64 * LP];

  const int tid  = threadIdx.x;
  const int lane = tid & 31;
  const int wave = tid >> 5;
  const int m_blk = blockIdx.y * 64;
  const int n_blk = blockIdx.x * 64;
  const int wm = (wave & 3) * 16;              // wave's row offset in macro-tile
  const int wn = (wave >> 2) * 32;             // wave's col offset in macro-tile
  const int half = lane >> 4;                  // 0: lanes 0-15, 1: lanes 16-31
  const int l16  = lane & 15;

  v8f acc0 = {};
  v8f acc1 = {};

  // cooperative-load mapping: 64 rows x 32 cols = 512 float4, 2 per thread
  const int r0  = tid >> 3;                    // 0..31
  const int r1  = 32 + r0;                     // 32..63
  const int kq  = (tid & 7) << 2;              // 0,4,...,28

  for (int k0 = 0; k0 < SD; k0 += 32) {
    *(float4*)&As[r0 * LP + kq] = *(const float4*)&A[(size_t)(m_blk + r0) * SD + k0 + kq];
    *(float4*)&As[r1 * LP + kq] = *(const float4*)&A[(size_t)(m_blk + r1) * SD + k0 + kq];
    *(float4*)&Bs[r0 * LP + kq] = *(const float4*)&B[(size_t)(n_blk + r0) * SD + k0 + kq];
    *(float4*)&Bs[r1 * LP + kq] = *(const float4*)&B[(size_t)(n_blk + r1) * SD + k0 + kq];
    __syncthreads();

#pragma unroll
    for (int kk = 0; kk < 32; kk += 4) {
      // A fragment: lane holds A[m= wm+l16, k = kk+2*half + {0,1}]
      v2f a  = *(const v2f*)&As[(wm + l16) * LP + kk + 2 * half];
      // B fragment: lane holds op(B)[k = kk+2*half + {0,1}, n = wn(+16)+l16]
      v2f b0 = *(const v2f*)&Bs[(wn + l16) * LP + kk + 2 * half];
      v2f b1 = *(const v2f*)&Bs[(wn + 16 + l16) * LP + kk + 2 * half];
      acc0 = __builtin_amdgcn_wmma_f32_16x16x4_f32(false, a, false, b0,
                                                   (short)0, acc0, false, false);
      acc1 = __builtin_amdgcn_wmma_f32_16x16x4_f32(false, a, false, b1,
                                                   (short)0, acc1, false, false);
    }
    __syncthreads();
  }

  // C/D layout: VGPR j -> rows j (lanes 0-15) and j+8 (lanes 16-31), col = l16
#pragma unroll
  for (int j = 0; j < 8; ++j) {
    const int row = m_blk + wm + j + 8 * half;
    C[(size_t)row * SD + n_blk + wn + l16]      = acc0[j];
    C[(size_t)row * SD + n_blk + wn + 16 + l16] = acc1[j];
  }
}

// ---------------------------------------------------------------------------
// In-place causal softmax over row s of SC (raw scores); scale = scale1*beta1.
// Row 0 is zeroed entirely (reference: A1[0,:]=0 post-softmax).
// ---------------------------------------------------------------------------
__global__ __launch_bounds__(256) void softmax_causal(
    float* __restrict__ SC, const float* __restrict__ beta1, float scale1) {
  const int s = blockIdx.x;
  float* row = SC + (size_t)s * SD;
  __shared__ float red[256];
  const int tid = threadIdx.x;

  if (s == 0) {
    for (int t = tid; t < SD; t += 256) row[t] = 0.0f;
    return;
  }
  const float sb = scale1 * beta1[0];
  const int n = s + 1;  // valid (unmasked) entries

  float mx = -__builtin_inff();
  for (int t = tid; t < n; t += 256) mx = fmaxf(mx, sb * row[t]);
  red[tid] = mx; __syncthreads();
  for (int off = 128; off > 0; off >>= 1) {
    if (tid < off) red[tid] = fmaxf(red[tid], red[tid + off]);
    __syncthreads();
  }
  mx = red[0]; __syncthreads();

  float sum = 0.0f;
  for (int t = tid; t < n; t += 256) {
    float e = expf(sb * row[t] - mx);
    row[t] = e;
    sum += e;
  }
  red[tid] = sum; __syncthreads();
  for (int off = 128; off > 0; off >>= 1) {
    if (tid < off) red[tid] += red[tid + off];
    __syncthreads();
  }
  const float inv = 1.0f / red[0];
  for (int t = tid; t < n; t += 256) row[t] *= inv;
  for (int t = n + tid; t < SD; t += 256) row[t] = 0.0f;
}

// ---------------------------------------------------------------------------
// out[b,v] = sum_t Wt[b*wstride + t] * Wemb[ids[b,t], v]
// (weighted sum of gathered embedding rows; Wemb is L2-resident)
// ---------------------------------------------------------------------------
__global__ __launch_bounds__(256) void weighted_embed_sum(
    const float* __restrict__ Wt, int wstride, const int* __restrict__ ids,
    const float* __restrict__ Wemb, float* __restrict__ out) {
  const int b = blockIdx.y;
  const int v = blockIdx.x * 256 + threadIdx.x;
  const float* w = Wt + (size_t)b * wstride;
  const int* idr = ids + (size_t)b * SD;
  float acc = 0.0f;
  for (int t = 0; t < SD; ++t) {
    acc = fmaf(w[t], Wemb[(size_t)idr[t] * SD + v], acc);
  }
  out[(size_t)b * SD + v] = acc;
}

// ---------------------------------------------------------------------------
// out[b,i] = alpha * (beta?*beta:1) * sum_k x[b,k]*W[i,k]   (+ optional Wemb
// row for the last token: Z1last = E_last + m @ WV1^T)
// ---------------------------------------------------------------------------
__global__ __launch_bounds__(256) void matvec_nt(
    const float* __restrict__ x, const float* __restrict__ W,
    float* __restrict__ out, const float* __restrict__ beta, float alpha,
    const float* __restrict__ addWemb, const int* __restrict__ ids) {
  const int b = blockIdx.y;
  const int i = blockIdx.x * 256 + threadIdx.x;
  const float* xr = x + (size_t)b * SD;
  const float* wr = W + (size_t)i * SD;
  float acc = 0.0f;
  for (int k = 0; k < SD; k += 4) {
    float4 wv = *(const float4*)&wr[k];
    float4 xv = *(const float4*)&xr[k];
    acc = fmaf(wv.x, xv.x, acc);
    acc = fmaf(wv.y, xv.y, acc);
    acc = fmaf(wv.z, xv.z, acc);
    acc = fmaf(wv.w, xv.w, acc);
  }
  float s = alpha * (beta ? beta[0] : 1.0f);
  float r = s * acc;
  if (addWemb) r += addWemb[(size_t)ids[(size_t)b * SD + (SD - 1)] * SD + i];
  out[(size_t)b * SD + i] = r;
}

// out[b,i] = sum_k x[b,k] * W[k,i]   (NN form: coalesced row streaming)
__global__ __launch_bounds__(256) void matvec_nn(
    const float* __restrict__ x, const float* __restrict__ W,
    float* __restrict__ out) {
  const int b = blockIdx.y;
  const int i = blockIdx.x * 256 + threadIdx.x;
  const float* xr = x + (size_t)b * SD;
  float acc = 0.0f;
  for (int k = 0; k < SD; ++k) acc = fmaf(xr[k], W[(size_t)k * SD + i], acc);
  out[(size_t)b * SD + i] = acc;
}

// ---------------------------------------------------------------------------
// c[b,t] = E[b,t]·g[b],  d[b,t] = E[b,t]·u[b]  (one wave per t, shared E row)
// ---------------------------------------------------------------------------
__global__ __launch_bounds__(256) void dots_cd(
    const int* __restrict__ ids, const float* __restrict__ Wemb,
    const float* __restrict__ g, const float* __restrict__ u,
    float* __restrict__ c, float* __restrict__ d) {
  const int b = blockIdx.y;
  const int wave = threadIdx.x >> 5, lane = threadIdx.x & 31;
  const int t = blockIdx.x * 8 + wave;
  const float* row = Wemb + (size_t)ids[(size_t)b * SD + t] * SD;
  const float* gb = g + (size_t)b * SD;
  const float* ub = u + (size_t)b * SD;
  float ac = 0.0f, ad = 0.0f;
  for (int k = lane; k < SD; k += 32) {
    float e = row[k];
    ac = fmaf(e, gb[k], ac);
    ad = fmaf(e, ub[k], ad);
  }
  for (int off = 16; off > 0; off >>= 1) {
    ac += __shfl_down(ac, off, 32);
    ad += __shfl_down(ad, off, 32);
  }
  if (lane == 0) {
    c[(size_t)b * SD + t] = ac;
    d[(size_t)b * SD + t] = ad;
  }
}

// ---------------------------------------------------------------------------
// s2[b,s] = scale2*beta2*( d[b,s] + sum_t A1[s,t]*c[b,t] );  s2[b,S-1] = -inf
// One wave per output s (coalesced A1 row reads; A1 rows already causal-zeroed)
// ---------------------------------------------------------------------------
__global__ __launch_bounds__(256) void s2_scores(
    const float* __restrict__ A1, const float* __restrict__ c,
    const float* __restrict__ d, float* __restrict__ s2,
    const float* __restrict__ beta2, float scale2) {
  const int b = blockIdx.y;
  const int wave = threadIdx.x >> 5, lane = threadIdx.x & 31;
  const int s = blockIdx.x * 8 + wave;
  const float* ar = A1 + (size_t)s * SD;
  const float* cb = c + (size_t)b * SD;
  float acc = 0.0f;
  for (int t = lane; t < SD; t += 32) acc = fmaf(ar[t], cb[t], acc);
  for (int off = 16; off > 0; off >>= 1) acc += __shfl_down(acc, off, 32);
  if (lane == 0) {
    float v = scale2 * beta2[0] * (d[(size_t)b * SD + s] + acc);
    s2[(size_t)b * SD + s] = (s == SD - 1) ? -__builtin_inff() : v;
  }
}

// In-place softmax over s-dim of s2 (one block per batch); -inf -> 0 weight.
__global__ __launch_bounds__(256) void softmax_s2(float* __restrict__ s2) {
  const int b = blockIdx.x;
  float* row = s2 + (size_t)b * SD;
  __shared__ float red[256];
  const int tid = threadIdx.x;
  float mx = -__builtin_inff();
  for (int t = tid; t < SD; t += 256) mx = fmaxf(mx, row[t]);
  red[tid] = mx; __syncthreads();
  for (int off = 128; off > 0; off >>= 1) {
    if (tid < off) red[tid] = fmaxf(red[tid], red[tid + off]);
    __syncthreads();
  }
  mx = red[0]; __syncthreads();
  float sum = 0.0f;
  for (int t = tid; t < SD; t += 256) {
    float e = expf(row[t] - mx);
    row[t] = e;
    sum += e;
  }
  red[tid] = sum; __syncthreads();
  for (int off = 128; off > 0; off >>= 1) {
    if (tid < off) red[tid] += red[tid + off];
    __syncthreads();
  }
  const float inv = 1.0f / red[0];
  for (int t = tid; t < SD; t += 256) row[t] *= inv;
}

// ---------------------------------------------------------------------------
extern "C" void kernel_launch(void* const* d_in, const int* in_sizes, int n_in,
                              void* d_out, int out_size, void* d_ws,
                              size_t ws_size, hipStream_t stream) {
  (void)in_sizes; (void)n_in; (void)out_size; (void)ws_size;
  const int*   ids   = (const int*)  d_in[0];
  const float* Wemb  = (const float*)d_in[1];
  const float* Ppos  = (const float*)d_in[2];
  const float* WQ1   = (const float*)d_in[3];
  const float* WK1   = (const float*)d_in[4];
  const float* WV1   = (const float*)d_in[5];
  const float* WQ2   = (const float*)d_in[6];
  const float* WK2   = (const float*)d_in[7];
  const float* WV2   = (const float*)d_in[8];
  const float* beta1 = (const float*)d_in[9];
  const float* beta2 = (const float*)d_in[10];
  const float* betao = (const float*)d_in[11];
  float* out = (float*)d_out;

  // workspace layout (f32): 3x S*S + 10x B*S  ~= 49.3 MB
  float* Q1 = (float*)d_ws;
  float* K1 = Q1 + (size_t)SD * SD;
  float* A1 = K1 + (size_t)SD * SD;          // raw S1, softmaxed in place
  float* sm = A1 + (size_t)SD * SD;          // m[b,v]
  float* zq = sm + (size_t)BB * SD;          // Z1 last row
  float* q2 = zq + (size_t)BB * SD;
  float* uu = q2 + (size_t)BB * SD;          // u = q2 @ WK2
  float* gg = uu + (size_t)BB * SD;          // g = u @ WV1
  float* cc = gg + (size_t)BB * SD;          // c[b,t]
  float* dd = cc + (size_t)BB * SD;          // d[b,t]
  float* a2 = dd + (size_t)BB * SD;          // S2 -> A2 in place
  float* e2 = a2 + (size_t)BB * SD;
  float* y2 = e2 + (size_t)BB * SD;

  const float scale1 = sqrtf(2048.0f) * 4.0f;               // sqrt(S)*(D/S)^2
  const float scale2 = sqrtf(2048.0f) * 4.0f;               // sqrt(S)*(D/V)^2
  const float scaleo = sqrtf(2048.0f) * 2.82842712474619f;  // sqrt(V)*(D/V)^1.5

  const dim3 g64(SD / 64, SD / 64), t256(256);
  const dim3 gBV(SD / 256, BB), gW8(SD / 8, BB);

  // stage 1 score pipeline (WMMA f32 GEMMs)
  hipLaunchKernelGGL(gemm_nt_wmma_f32, g64, t256, 0, stream, Ppos, WQ1, Q1);
  hipLaunchKernelGGL(gemm_nt_wmma_f32, g64, t256, 0, stream, Ppos, WK1, K1);
  hipLaunchKernelGGL(gemm_nt_wmma_f32, g64, t256, 0, stream, Q1, K1, A1);
  hipLaunchKernelGGL(softmax_causal, dim3(SD), t256, 0, stream, A1, beta1, scale1);

  // last-row residual path: m -> Z1last -> q2 -> u -> g
  hipLaunchKernelGGL(weighted_embed_sum, gBV, t256, 0, stream,
                     A1 + (size_t)(SD - 1) * SD, 0, ids, Wemb, sm);
  hipLaunchKernelGGL(matvec_nt, gBV, t256, 0, stream,
                     sm, WV1, zq, (const float*)nullptr, 1.0f, Wemb, ids);
  hipLaunchKernelGGL(matvec_nt, gBV, t256, 0, stream,
                     zq, WQ2, q2, (const float*)nullptr, 1.0f,
                     (const float*)nullptr, ids);
  hipLaunchKernelGGL(matvec_nn, gBV, t256, 0, stream, q2, WK2, uu);
  hipLaunchKernelGGL(matvec_nn, gBV, t256, 0, stream, uu, WV1, gg);

  // stage 2 scores, softmax, output
  hipLaunchKernelGGL(dots_cd, gW8, t256, 0, stream, ids, Wemb, gg, uu, cc, dd);
  hipLaunchKernelGGL(s2_scores, gW8, t256, 0, stream, A1, cc, dd, a2, beta2, scale2);
  hipLaunchKernelGGL(softmax_s2, dim3(BB), t256, 0, stream, a2);
  hipLaunchKernelGGL(weighted_embed_sum, gBV, t256, 0, stream, a2, SD, ids, Wemb, e2);
  hipLaunchKernelGGL(matvec_nt, gBV, t256, 0, stream,
                     e2, WV2, y2, (const float*)nullptr, 1.0f,
                     (const float*)nullptr, ids);
  hipLaunchKernelGGL(matvec_nt, gBV, t256, 0, stream,
                     y2, Wemb, out, betao, scaleo, (const float*)nullptr, ids);
}